// agent_76879914598453
// MI455X (gfx1250) — compile-verified
//
#include <hip/hip_runtime.h>
#include <math.h>

typedef __attribute__((ext_vector_type(2))) float v2f;
typedef __attribute__((ext_vector_type(8))) float v8f;

#define NCENT 4096
#define MPTS  8192
#define DIMS  32
#define HID   100
#define TOT   (NCENT * DIMS)   // 131072

// workspace layout (float offsets)
#define WS_S      0                       // -0.5/var
#define WS_BASE   1                       // -0.5*dims*log(2*pi*var)
#define WS_PSUM   16                      // 256 partial sums
#define WS_PSQ    (16 + 256)              // 256 partial sumsq
#define WS_LOC    1024                    // loc[N][32]
#define WS_LOCSQ  (WS_LOC + NCENT * DIMS) // ||loc_n||^2
#define WS_XSQ    (WS_LOCSQ + NCENT)      // ||x_m||^2

// ---------- deterministic variance reduction (stage 1) ----------
__global__ __launch_bounds__(256) void k_var_partial(const float* __restrict__ cent,
                                                     float* __restrict__ ws) {
    __shared__ float ss[256];
    __shared__ float sq[256];
    int t = threadIdx.x;
    int base = blockIdx.x * 512;
    float a = cent[base + t];
    float b = cent[base + 256 + t];
    ss[t] = a + b;
    sq[t] = a * a + b * b;
    __syncthreads();
    for (int off = 128; off > 0; off >>= 1) {
        if (t < off) { ss[t] += ss[t + off]; sq[t] += sq[t + off]; }
        __syncthreads();
    }
    if (t == 0) {
        ws[WS_PSUM + blockIdx.x] = ss[0];
        ws[WS_PSQ  + blockIdx.x] = sq[0];
    }
}

// ---------- variance reduction (stage 2) ----------
__global__ __launch_bounds__(256) void k_var_final(float* __restrict__ ws) {
    __shared__ float ss[256];
    __shared__ float sq[256];
    int t = threadIdx.x;
    ss[t] = ws[WS_PSUM + t];
    sq[t] = ws[WS_PSQ + t];
    __syncthreads();
    for (int off = 128; off > 0; off >>= 1) {
        if (t < off) { ss[t] += ss[t + off]; sq[t] += sq[t + off]; }
        __syncthreads();
    }
    if (t == 0) {
        float cnt = (float)TOT;
        float var = (sq[0] - ss[0] * ss[0] / cnt) / (cnt - 1.0f);  // ddof=1
        ws[WS_S]    = -0.5f / var;
        ws[WS_BASE] = -0.5f * (float)DIMS * logf(6.283185307179586f * var);
    }
}

// ---------- small MLP: loc = tanh((c @ W1^T) @ W2^T + b2) * c ----------
__global__ __launch_bounds__(128) void k_mlp(const float* __restrict__ cent,
                                             const float* __restrict__ W1,
                                             const float* __restrict__ W2,
                                             const float* __restrict__ b2,
                                             float* __restrict__ ws) {
    __shared__ float sc[DIMS];
    __shared__ float sh[HID];
    __shared__ float sl[DIMS];
    int n = blockIdx.x;
    int t = threadIdx.x;
    if (t < DIMS) sc[t] = cent[n * DIMS + t];
    __syncthreads();
    if (t < HID) {
        float acc = 0.0f;
        const float* w = W1 + t * DIMS;   // W1[h, :]
        #pragma unroll
        for (int d = 0; d < DIMS; ++d) acc += sc[d] * w[d];
        sh[t] = acc;
    }
    __syncthreads();
    if (t < DIMS) {
        float acc = b2[t];
        const float* w = W2 + t * HID;    // W2[d, :]
        for (int h = 0; h < HID; ++h) acc += sh[h] * w[h];
        float lv = tanhf(acc) * sc[t];
        ws[WS_LOC + n * DIMS + t] = lv;
        sl[t] = lv * lv;
    }
    __syncthreads();
    if (t == 0) {
        float a = 0.0f;
        #pragma unroll
        for (int d = 0; d < DIMS; ++d) a += sl[d];
        ws[WS_LOCSQ + n] = a;
    }
}

// ---------- ||x_m||^2 ----------
__global__ __launch_bounds__(256) void k_xsq(const float* __restrict__ x,
                                             float* __restrict__ ws) {
    int m = blockIdx.x * 256 + threadIdx.x;
    const float* r = x + (size_t)m * DIMS;
    float a = 0.0f;
    #pragma unroll
    for (int d = 0; d < DIMS; ++d) a += r[d] * r[d];
    ws[WS_XSQ + m] = a;
}

// ---------- main pairwise kernel: one wave per 16x16 output tile ----------
// out[m, n] = exp( s * (xsq[m] + locsq[n] - 2 * (x_m . loc_n)) + base )
__global__ __launch_bounds__(256) void k_pairwise(const float* __restrict__ X,
                                                  const float* __restrict__ ws,
                                                  float* __restrict__ out) {
    const float* loc   = ws + WS_LOC;
    const float* locsq = ws + WS_LOCSQ;
    const float* xsq   = ws + WS_XSQ;

    int lane = threadIdx.x & 31;
    int wave = threadIdx.x >> 5;
    int tile = blockIdx.x * 8 + wave;
    int tn = tile & (NCENT / 16 - 1);   // 256 column tiles
    int tm = tile >> 8;                 // 512 row tiles

    int half = lane >> 4;               // 0: K=0,1  1: K=2,3 (A 16x4 f32 layout)
    int l15  = lane & 15;
    int kb   = half * 2;

    const float* xrow = X   + (size_t)(tm * 16 + l15) * DIMS;  // A: M = l15
    const float* lrow = loc + (size_t)(tn * 16 + l15) * DIMS;  // B: N = l15 (B[k][n]=loc[n][k])

    v8f acc = {};
    #pragma unroll
    for (int kk = 0; kk < 8; ++kk) {
        v2f a = *(const v2f*)(xrow + kk * 4 + kb);
        v2f b = *(const v2f*)(lrow + kk * 4 + kb);
        acc = __builtin_amdgcn_wmma_f32_16x16x4_f32(
            /*neg_a=*/false, a, /*neg_b=*/false, b,
            /*c_mod=*/(short)0, acc, /*reuse_a=*/false, /*reuse_b=*/false);
    }

    float s    = ws[WS_S];
    float base = ws[WS_BASE];
    int n_g = tn * 16 + l15;
    float lq = locsq[n_g];
    int m0 = tm * 16 + half * 8;        // C/D layout: vgpr v, upper half -> M = v + 8
    #pragma unroll
    for (int v = 0; v < 8; ++v) {
        int m_g = m0 + v;
        float sqd = xsq[m_g] + lq - 2.0f * acc[v];
        out[(size_t)m_g * NCENT + n_g] = expf(fmaf(s, sqd, base));
    }
}

extern "C" void kernel_launch(void* const* d_in, const int* in_sizes, int n_in,
                              void* d_out, int out_size, void* d_ws, size_t ws_size,
                              hipStream_t stream) {
    const float* cent = (const float*)d_in[0];  // [4096, 32]
    const float* x    = (const float*)d_in[1];  // [8192, 32]
    const float* W1   = (const float*)d_in[2];  // [100, 32]
    const float* W2   = (const float*)d_in[3];  // [32, 100]
    const float* b2   = (const float*)d_in[4];  // [32]
    float* out = (float*)d_out;                 // [8192, 4096]
    float* ws  = (float*)d_ws;

    k_var_partial<<<256, 256, 0, stream>>>(cent, ws);
    k_var_final<<<1, 256, 0, stream>>>(ws);
    k_mlp<<<NCENT, 128, 0, stream>>>(cent, W1, W2, b2, ws);
    k_xsq<<<MPTS / 256, 256, 0, stream>>>(x, ws);
    k_pairwise<<<(MPTS / 16) * (NCENT / 16) / 8, 256, 0, stream>>>(x, ws, out);
}